// ChebNet_52750788329578
// MI455X (gfx1250) — compile-verified
//
#include <hip/hip_runtime.h>
#include <math.h>

#define Bsz   2048
#define BANDS 5
#define NN    30
#define INCC  5
#define FF    64
#define KORD  4
#define OUTD  9

typedef __attribute__((ext_vector_type(2))) float v2f;
typedef __attribute__((ext_vector_type(8))) float v8f;

// ---- f32 WMMA helpers (V_WMMA_F32_16X16X4_F32, wave32) --------------------
__device__ __forceinline__ v8f wmma_f32(v2f a, v2f b, v8f c) {
  return __builtin_amdgcn_wmma_f32_16x16x4_f32(false, a, false, b, (short)0, c,
                                               false, false);
}

// A fragment: 16x4 (M x K). Lanes 0-15: K = k0+0 / k0+1 in VGPR0/1;
// lanes 16-31: K = k0+2 / k0+3.  m = mBase + lane%16.
__device__ __forceinline__ v2f fragA(const float* M, int stride, int mBase,
                                     int k0, int lane) {
  int m  = mBase + (lane & 15);
  int ko = k0 + ((lane & 16) ? 2 : 0);
  v2f a;
  a.x = M[m * stride + ko];
  a.y = M[m * stride + ko + 1];
  return a;
}

// B fragment: 4x16 (K x N). Row-striped across lanes within each VGPR:
// VGPR0: K=k0 (lanes 0-15) / K=k0+2 (lanes 16-31); VGPR1: k0+1 / k0+3.
__device__ __forceinline__ v2f fragB(const float* M, int stride, int k0,
                                     int nBase, int lane) {
  int n  = nBase + (lane & 15);
  int ko = k0 + ((lane & 16) ? 2 : 0);
  v2f b;
  b.x = M[ko * stride + n];
  b.y = M[(ko + 1) * stride + n];
  return b;
}

// C/D layout: VGPR v, lanes 0-15 -> (M=v,    N=lane),
//                     lanes 16-31 -> (M=v+8, N=lane-16)
__device__ __forceinline__ void storeC(float* M, int stride, int mBase,
                                       int nBase, int lane, v8f c) {
  int col = nBase + (lane & 15);
  int rb  = mBase + ((lane & 16) ? 8 : 0);
#pragma unroll
  for (int v = 0; v < 8; ++v) M[(rb + v) * stride + col] = c[v];
}

__device__ __forceinline__ v8f loadC(const float* M, int stride, int mBase,
                                     int nBase, int lane) {
  int col = nBase + (lane & 15);
  int rb  = mBase + ((lane & 16) ? 8 : 0);
  v8f c;
#pragma unroll
  for (int v = 0; v < 8; ++v) c[v] = M[(rb + v) * stride + col];
  return c;
}

// ===========================================================================
// Kernel 1: per (b,band) fused Laplacian + Cheb basis + ChebConv x2 + CVLinear
// ===========================================================================
__global__ __launch_bounds__(256) void cheb_fused(
    const float* __restrict__ real, const float* __restrict__ imag,
    const float* __restrict__ graph, const float* __restrict__ W1,
    const float* __restrict__ b1, const float* __restrict__ W2,
    const float* __restrict__ b2, const float* __restrict__ linWr,
    const float* __restrict__ linWi, const float* __restrict__ linbr,
    const float* __restrict__ linbi, float* __restrict__ wsR,
    float* __restrict__ wsI) {
  // LDS map (floats):
  //   sT   : 3 * 2 * (32*33)  = 6336   (T1..T3 real/imag, stride 33, padded)
  //   sX1r/i: 2 * (32*65)     = 4160   (conv1 output, stride 65)
  //   sZr/i : 2 * (32*65)     = 4160   (A staging / Z1 / Z_k / y staging)
  //   sXr/i : 2 * 160                  (input x, 30x5)
  //   sDinv : 32
  __shared__ float smem[15008];
  float* sT    = smem;              // [3][2][1056]
  float* sX1r  = smem + 6336;
  float* sX1i  = sX1r + 2080;
  float* sZr   = sX1i + 2080;
  float* sZi   = sZr + 2080;
  float* sXr   = sZi + 2080;
  float* sXi   = sXr + 160;
  float* sDinv = sXi + 160;

  const int tid  = threadIdx.x;
  const int lane = tid & 31;
  const int wv   = tid >> 5;
  const int pair = blockIdx.x;
  const int b    = pair / BANDS;

  for (int i = tid; i < 15008; i += 256) smem[i] = 0.0f;
  __syncthreads();

  // ---- load graph tile (into sZr, stride 33) and inputs -------------------
  const float* A = graph + (size_t)pair * (NN * NN);
  for (int idx = tid; idx < NN * NN; idx += 256)
    sZr[(idx / NN) * 33 + (idx % NN)] = A[idx];
  for (int idx = tid; idx < NN * INCC; idx += 256) {
    sXr[idx] = real[(size_t)b * NN * INCC + idx];
    sXi[idx] = imag[(size_t)b * NN * INCC + idx];
  }
  __syncthreads();

  // ---- dinv ---------------------------------------------------------------
  if (tid < NN) {
    float s = 0.0f;
    for (int j = 0; j < NN; ++j)
      s += 0.5f * (sZr[tid * 33 + j] + sZr[j * 33 + tid]) +
           ((j == tid) ? 1.0f : 0.0f);
    sDinv[tid] = 1.0f / sqrtf(s);
  }
  __syncthreads();

  // ---- magnetic Laplacian -> T1 (sT[0]) -----------------------------------
  float* T1r = sT;
  float* T1i = sT + 1056;
  for (int idx = tid; idx < NN * NN; idx += 256) {
    int i = idx / NN, j = idx % NN;
    float a   = sZr[i * 33 + j];
    float at  = sZr[j * 33 + i];
    float as_ = 0.5f * (a + at) + ((i == j) ? 1.0f : 0.0f);
    float na  = sDinv[i] * sDinv[j] * as_;
    float th  = 1.5707963267948966f * (a - at);  // 2*pi*q, q=0.25
    float sn, cs;
    __sincosf(th, &sn, &cs);
    T1r[i * 33 + j] = -na * cs;
    T1i[i * 33 + j] = -na * sn;
  }
  __syncthreads();

  // ---- Chebyshev recurrence: T2, T3 via f32 WMMA --------------------------
  {
    const int part = wv >> 2;  // 0 = real, 1 = imag job for this wave
    const int mB   = ((wv >> 1) & 1) * 16;
    const int nB   = (wv & 1) * 16;

#pragma unroll
    for (int s = 2; s <= 3; ++s) {
      const float* Tpr = (s == 2) ? T1r : (sT + 2112);
      const float* Tpi = (s == 2) ? T1i : (sT + 2112 + 1056);
      v8f acc = {};
#pragma unroll
      for (int kk = 0; kk < 8; ++kk) {
        int k0 = kk * 4;
        v2f aR = fragA(T1r, 33, mB, k0, lane);
        v2f aI = fragA(T1i, 33, mB, k0, lane);
        v2f bR = fragB(Tpr, 33, k0, nB, lane);
        v2f bI = fragB(Tpi, 33, k0, nB, lane);
        // real: acc += Lr@Tr + (-Li)@Ti ; imag: acc += Lr@Ti + Li@Tr
        v2f bFirst, bSecond, aSecond;
        if (part == 0) {
          bFirst  = bR;
          bSecond = bI;
          aSecond.x = -aI.x;
          aSecond.y = -aI.y;
        } else {
          bFirst  = bI;
          bSecond = bR;
          aSecond = aI;
        }
        acc = wmma_f32(aR, bFirst, acc);
        acc = wmma_f32(aSecond, bSecond, acc);
      }
      float* dst = sT + (s - 1) * 2112 + part * 1056;
      int col = nB + (lane & 15);
      int rb  = mB + ((lane & 16) ? 8 : 0);
      if (s == 2) {
        // T2 = 2*L@T1 - I  (identity only on 30 valid diagonal entries)
#pragma unroll
        for (int v = 0; v < 8; ++v) {
          float r   = 2.0f * acc[v];
          int   row = rb + v;
          if (part == 0 && row == col && row < NN) r -= 1.0f;
          dst[row * 33 + col] = r;
        }
      } else {
        v8f pp = loadC(sT + part * 1056, 33, mB, nB, lane);  // T1
#pragma unroll
        for (int v = 0; v < 8; ++v)
          dst[(rb + v) * 33 + col] = 2.0f * acc[v] - pp[v];
      }
      __syncthreads();
    }
  }

  // ---- ChebConv1 (C=5, tiny -> VALU).  Z1[k] for k=1..3 (k=0 is x). -------
  for (int idx = tid; idx < 3 * NN * INCC; idx += 256) {
    int k1  = idx / (NN * INCC);  // 0..2  -> order k = k1+1
    int rem = idx % (NN * INCC);
    int m = rem / INCC, c = rem % INCC;
    const float* Tr = sT + k1 * 2112;
    const float* Ti = Tr + 1056;
    float zr = 0.0f, zi = 0.0f;
    for (int n = 0; n < NN; ++n) {
      float tr = Tr[m * 33 + n], ti = Ti[m * 33 + n];
      float xr = sXr[n * INCC + c], xi = sXi[n * INCC + c];
      zr += tr * xr - ti * xi;
      zi += tr * xi + ti * xr;
    }
    sZr[idx] = zr;
    sZi[idx] = zi;
  }
  __syncthreads();

  // x1[m][f] = b1[f] + sum_k sum_c Z1[k][m][c] * W1[k][c][f]   (W,b real)
  for (int idx = tid; idx < NN * FF; idx += 256) {
    int m = idx >> 6, f = idx & 63;
    float bias = b1[f];
    float ar = bias, ai = bias;
#pragma unroll
    for (int k = 0; k < KORD; ++k)
      for (int c = 0; c < INCC; ++c) {
        float w = W1[(k * INCC + c) * FF + f];
        float zr, zi;
        if (k == 0) {
          zr = sXr[m * INCC + c];
          zi = sXi[m * INCC + c];
        } else {
          int zidx = ((k - 1) * NN + m) * INCC + c;
          zr = sZr[zidx];
          zi = sZi[zidx];
        }
        ar += zr * w;
        ai += zi * w;
      }
    sX1r[m * 65 + f] = ar;
    sX1i[m * 65 + f] = ai;
  }
  __syncthreads();

  // ---- ChebConv2 via f32 WMMA ---------------------------------------------
  {
    const int mB = (wv >> 2) * 16;  // row tile   (0 or 16)
    const int fB = (wv & 3) * 16;   // col tile   (0,16,32,48)
    v8f yr = {}, yi = {};
    for (int k = 0; k < KORD; ++k) {
      if (k > 0) {
        __syncthreads();
        const float* Tr = sT + (k - 1) * 2112;
        const float* Ti = Tr + 1056;
        v8f zr = {}, zi = {};
#pragma unroll
        for (int kk = 0; kk < 8; ++kk) {
          int k0 = kk * 4;
          v2f aR = fragA(Tr, 33, mB, k0, lane);
          v2f aI = fragA(Ti, 33, mB, k0, lane);
          v2f bR = fragB(sX1r, 65, k0, fB, lane);
          v2f bI = fragB(sX1i, 65, k0, fB, lane);
          v2f nI;
          nI.x = -aI.x;
          nI.y = -aI.y;
          zr = wmma_f32(aR, bR, zr);
          zr = wmma_f32(nI, bI, zr);
          zi = wmma_f32(aR, bI, zi);
          zi = wmma_f32(aI, bR, zi);
        }
        storeC(sZr, 65, mB, fB, lane, zr);
        storeC(sZi, 65, mB, fB, lane, zi);
        __syncthreads();
      }
      const float* Ar = (k == 0) ? sX1r : sZr;
      const float* Ai = (k == 0) ? sX1i : sZi;
      const float* Wk = W2 + k * FF * FF;
      const int f  = fB + (lane & 15);
#pragma unroll
      for (int kk = 0; kk < 16; ++kk) {
        int c0 = kk * 4;
        v2f aR = fragA(Ar, 65, mB, c0, lane);
        v2f aI = fragA(Ai, 65, mB, c0, lane);
        int co = c0 + ((lane & 16) ? 2 : 0);
        v2f bw;
        bw.x = Wk[co * FF + f];
        bw.y = Wk[(co + 1) * FF + f];
        yr = wmma_f32(aR, bw, yr);
        yi = wmma_f32(aI, bw, yi);
      }
    }
    __syncthreads();
    float bb = b2[fB + (lane & 15)];
#pragma unroll
    for (int v = 0; v < 8; ++v) {
      yr[v] += bb;
      yi[v] += bb;
    }
    storeC(sZr, 65, mB, fB, lane, yr);
    storeC(sZi, 65, mB, fB, lane, yi);
    __syncthreads();
  }

  // ---- CVLinear F -> 1 ----------------------------------------------------
  if (tid < NN) {
    float lr = linbr[0], li = linbi[0];
    for (int f = 0; f < FF; ++f) {
      float ar = sZr[tid * 65 + f], ai = sZi[tid * 65 + f];
      float wr = linWr[f], wi = linWi[f];
      lr += ar * wr - ai * wi;
      li += ar * wi + ai * wr;
    }
    wsR[(size_t)pair * NN + tid] = lr;
    wsI[(size_t)pair * NN + tid] = li;
  }
}

// ===========================================================================
// Kernel 2: complex BN statistics over the batch axis, per (band, node)
// ===========================================================================
__global__ __launch_bounds__(256) void bn_stats(const float* __restrict__ wsR,
                                                const float* __restrict__ wsI,
                                                float* __restrict__ bnp) {
  __shared__ float red[5][256];
  const int t    = threadIdx.x;
  const int band = blockIdx.x / NN;
  const int n    = blockIdx.x % NN;
  float sr = 0, si = 0, srr = 0, sii = 0, sri = 0;
  for (int b = t; b < Bsz; b += 256) {
    size_t idx = (size_t)(b * BANDS + band) * NN + n;
    float r = wsR[idx], i = wsI[idx];
    sr += r;
    si += i;
    srr += r * r;
    sii += i * i;
    sri += r * i;
  }
  red[0][t] = sr;
  red[1][t] = si;
  red[2][t] = srr;
  red[3][t] = sii;
  red[4][t] = sri;
  __syncthreads();
  for (int s = 128; s > 0; s >>= 1) {
    if (t < s)
      for (int q = 0; q < 5; ++q) red[q][t] += red[q][t + s];
    __syncthreads();
  }
  if (t == 0) {
    const float inv = 1.0f / (float)Bsz;
    float mr  = red[0][0] * inv, mi = red[1][0] * inv;
    float Vrr = red[2][0] * inv - mr * mr + 1e-5f;
    float Vii = red[3][0] * inv - mi * mi + 1e-5f;
    float Vri = red[4][0] * inv - mr * mi;
    float s_  = sqrtf(Vrr * Vii - Vri * Vri);
    float tt  = sqrtf(Vrr + Vii + 2.0f * s_);
    float ist = 1.0f / (s_ * tt);
    float* p  = bnp + blockIdx.x * 5;
    p[0] = mr;
    p[1] = mi;
    p[2] = (Vii + s_) * ist;   // Wrr
    p[3] = -Vri * ist;         // Wri
    p[4] = (Vrr + s_) * ist;   // Wii
  }
}

// ===========================================================================
// Kernel 3: apply BN affine + polar tanh + CVLinear 30 -> 9
// ===========================================================================
__global__ __launch_bounds__(32) void bn_tanh_fc(
    const float* __restrict__ wsR, const float* __restrict__ wsI,
    const float* __restrict__ bnp, const float* __restrict__ grr,
    const float* __restrict__ gri, const float* __restrict__ gii,
    const float* __restrict__ btr, const float* __restrict__ bti,
    const float* __restrict__ fcWr, const float* __restrict__ fcWi,
    const float* __restrict__ fcbr, const float* __restrict__ fcbi,
    float* __restrict__ out) {
  __shared__ float xr[NN], xi[NN];
  const int t    = threadIdx.x;
  const int pair = blockIdx.x;
  const int band = pair % BANDS;
  if (t < NN) {
    const float* p = bnp + (band * NN + t) * 5;
    float r  = wsR[(size_t)pair * NN + t] - p[0];
    float i  = wsI[(size_t)pair * NN + t] - p[1];
    float hr = p[2] * r + p[3] * i;
    float hi = p[3] * r + p[4] * i;
    float vr = grr[t] * hr + gri[t] * hi + btr[t];
    float vi = gri[t] * hr + gii[t] * hi + bti[t];
    float mag = sqrtf(vr * vr + vi * vi);
    float sc  = tanhf(mag) / fmaxf(mag, 1e-12f);
    xr[t] = vr * sc;
    xi[t] = vi * sc;
  }
  __syncthreads();
  if (t < OUTD) {
    float orr = fcbr[t], oii = fcbi[t];
    for (int n = 0; n < NN; ++n) {
      float wr = fcWr[n * OUTD + t], wi = fcWi[n * OUTD + t];
      orr += xr[n] * wr - xi[n] * wi;
      oii += xr[n] * wi + xi[n] * wr;
    }
    out[((size_t)pair * OUTD + t) * 2 + 0] = orr;
    out[((size_t)pair * OUTD + t) * 2 + 1] = oii;
  }
}

// ===========================================================================
extern "C" void kernel_launch(void* const* d_in, const int* in_sizes, int n_in,
                              void* d_out, int out_size, void* d_ws,
                              size_t ws_size, hipStream_t stream) {
  const float* real  = (const float*)d_in[0];
  const float* imag  = (const float*)d_in[1];
  const float* graph = (const float*)d_in[2];
  // d_in[3] = layer (fixed at 2 in the reference setup -> conv2 path taken)
  const float* W1    = (const float*)d_in[4];
  const float* b1    = (const float*)d_in[5];
  const float* W2    = (const float*)d_in[6];
  const float* b2    = (const float*)d_in[7];
  const float* linWr = (const float*)d_in[8];
  const float* linWi = (const float*)d_in[9];
  const float* linbr = (const float*)d_in[10];
  const float* linbi = (const float*)d_in[11];
  const float* grr   = (const float*)d_in[12];
  const float* gri   = (const float*)d_in[13];
  const float* gii   = (const float*)d_in[14];
  const float* btr   = (const float*)d_in[15];
  const float* bti   = (const float*)d_in[16];
  const float* fcWr  = (const float*)d_in[17];
  const float* fcWi  = (const float*)d_in[18];
  const float* fcbr  = (const float*)d_in[19];
  const float* fcbi  = (const float*)d_in[20];

  float* wsR = (float*)d_ws;
  float* wsI = wsR + (size_t)Bsz * BANDS * NN;
  float* bnp = wsI + (size_t)Bsz * BANDS * NN;

  cheb_fused<<<Bsz * BANDS, 256, 0, stream>>>(real, imag, graph, W1, b1, W2,
                                              b2, linWr, linWi, linbr, linbi,
                                              wsR, wsI);
  bn_stats<<<BANDS * NN, 256, 0, stream>>>(wsR, wsI, bnp);
  bn_tanh_fc<<<Bsz * BANDS, 32, 0, stream>>>(wsR, wsI, bnp, grr, gri, gii, btr,
                                             bti, fcWr, fcWi, fcbr, fcbi,
                                             (float*)d_out);
}